// _ECELoss_14216341750010
// MI455X (gfx1250) — compile-verified
//
#include <hip/hip_runtime.h>
#include <stdint.h>

#define TILE   2048          // elements per TDM tile (2048*12B = 24 KB of logits)
#define GRID   2048
#define BLOCK  256           // 8 wave32
#define NBINS  15

typedef unsigned int uint;
typedef uint  v4u __attribute__((ext_vector_type(4)));
typedef int   v4i __attribute__((ext_vector_type(4)));
typedef int   v8i __attribute__((ext_vector_type(8)));
typedef float v2f __attribute__((ext_vector_type(2)));
typedef float v8f __attribute__((ext_vector_type(8)));

#if defined(__AMDGCN__) && __has_builtin(__builtin_amdgcn_tensor_load_to_lds)
#define HAS_TDM 1
#else
#define HAS_TDM 0
#endif
#if defined(__AMDGCN__) && __has_builtin(__builtin_amdgcn_wmma_f32_16x16x4_f32)
#define HAS_WMMA4 1
#else
#define HAS_WMMA4 0
#endif

#if HAS_TDM
// Issue one TDM load of TILE*3 f32 from gsrc into LDS offset lds_off.
// D# built per CDNA5 ISA §8 (group0 128b, group1 256b), 1-D tile.
__device__ __forceinline__ void tdm_issue(const float* gsrc, uint lds_off) {
  uint64_t ga = (uint64_t)(uintptr_t)gsrc;
  const uint TD0 = TILE * 3;                     // 6144 f32 elements
  v4u g0;
  g0.x = 1u;                                     // count=1, is_restore=0, gather off
  g0.y = lds_off;                                // lds_addr (bytes)
  g0.z = (uint)(ga & 0xffffffffu);               // global_addr[31:0]
  g0.w = (uint)((ga >> 32) & 0x01ffffffu)        // global_addr[56:32]
       | (2u << 30);                             // type=2 ("image")
  v8i g1;
  g1[0] = (int)(2u << 16);                       // workgroup_mask=0, data_size=2 (4B)
  g1[1] = (int)((TD0 & 0xffffu) << 16);          // tensor_dim0[15:0] @ bits[63:48]
  g1[2] = (int)((TD0 >> 16) | (1u << 16));       // tensor_dim0[31:16] | tensor_dim1=1
  g1[3] = (int)(TD0 << 16);                      // tensor_dim1 hi=0 | tile_dim0=TD0
  g1[4] = 1;                                     // tile_dim1=1, tile_dim2=0
  g1[5] = (int)TD0;                              // tensor_dim0_stride[31:0]
  g1[6] = (int)((TD0 & 0xffffu) << 16);          // stride0 hi=0 | stride1[15:0]
  g1[7] = 0;                                     // stride1[47:16]
  v4i z4 = {0, 0, 0, 0};
#if defined(__clang_major__) && (__clang_major__ >= 23)
  v8i z8 = {0, 0, 0, 0, 0, 0, 0, 0};
  __builtin_amdgcn_tensor_load_to_lds(g0, g1, z4, z4, z8, 0);
#else
  __builtin_amdgcn_tensor_load_to_lds(g0, g1, z4, z4, 0);
#endif
}
#endif  // HAS_TDM

__global__ __launch_bounds__(BLOCK)
void ece_hist(const float* __restrict__ logits,
              const int*   __restrict__ lab32,    // int64 labels, read low dword
              float*       __restrict__ gbins,    // [48]: cnt | conf | acc
              int nElem, int nTiles) {
#if HAS_TDM
  __shared__ float tile[2][TILE * 3];
#endif
  __shared__ float hist[8][48];                   // per-wave privatized bins

  const int tid  = (int)threadIdx.x;
  const int wave = tid >> 5;

  for (int j = tid; j < 8 * 48; j += BLOCK) ((float*)hist)[j] = 0.0f;

  int t   = (int)blockIdx.x;
  int buf = 0;
#if HAS_TDM
  if (tid < 32 && t < nTiles)
    tdm_issue(logits + (size_t)t * TILE * 3, (uint)(uintptr_t)&tile[0][0]);
#endif

  while (t < nTiles) {
    int tn = t + (int)gridDim.x;
#if HAS_TDM
    if (tid < 32) {
      if (tn < nTiles) {
        tdm_issue(logits + (size_t)tn * TILE * 3, (uint)(uintptr_t)&tile[buf ^ 1][0]);
        __builtin_amdgcn_s_wait_tensorcnt(1);     // tile t complete (in-order)
      } else {
        __builtin_amdgcn_s_wait_tensorcnt(0);
      }
    }
#endif
    __syncthreads();                              // tile t visible to all waves

    const int base = t * TILE;
#pragma unroll
    for (int k = 0; k < TILE / BLOCK; ++k) {
      int e   = tid + k * BLOCK;
      int idx = base + e;
      if (idx < nElem) {
#if HAS_TDM
        const float* lp = &tile[buf][e * 3];
#else
        const float* lp = logits + (size_t)idx * 3;
#endif
        float a0 = lp[0], a1 = lp[1], a2 = lp[2];
        float m    = fmaxf(a0, fmaxf(a1, a2));
        float z    = __expf(a0 - m) + __expf(a1 - m) + __expf(a2 - m);
        float conf = __builtin_amdgcn_rcpf(z);    // max prob = exp(0)/Z = 1/Z
        int   pred = (a1 > a0) ? ((a2 > a1) ? 2 : 1) : ((a2 > a0) ? 2 : 0);
        float acc  = (pred == lab32[2 * (size_t)idx]) ? 1.0f : 0.0f;
        int b = (int)ceilf(conf * 15.0f) - 1;     // searchsorted(side='left')
        b = b < 0 ? 0 : (b > NBINS - 1 ? NBINS - 1 : b);
        atomicAdd(&hist[wave][b],      1.0f);     // ds_add_f32, wave-private
        atomicAdd(&hist[wave][16 + b], conf);
        atomicAdd(&hist[wave][32 + b], acc);
      }
    }
    __syncthreads();                              // done reading buf before reuse
    t = tn;
    buf ^= 1;
  }

  __syncthreads();
#if HAS_WMMA4
  // Reduce 8 wave-hists -> 48 sums with v_wmma_f32_16x16x4_f32.
  // A = ones(16x4); B[k][n] = hist[kbase+k][chunk*16+n]; two WMMAs cover 8 waves.
  {
    int lane = tid & 31;
    int w    = tid >> 5;
    if (w < 3) {                                  // chunk 0:cnt 1:conf 2:acc
      int n     = lane & 15;
      int kbase = (lane < 16) ? 0 : 2;
      int col   = w * 16 + n;
      v2f a = {1.0f, 1.0f};
      v8f c = {};
      v2f b0 = {hist[kbase][col],     hist[kbase + 1][col]};
      c = __builtin_amdgcn_wmma_f32_16x16x4_f32(false, a, false, b0,
                                                (short)0, c, false, false);
      v2f b1 = {hist[kbase + 4][col], hist[kbase + 5][col]};
      c = __builtin_amdgcn_wmma_f32_16x16x4_f32(false, a, false, b1,
                                                (short)0, c, false, false);
      if (lane < 16) atomicAdd(&gbins[col], c[0]);
    }
  }
#else
  if (tid < 48) {
    float s = 0.0f;
#pragma unroll
    for (int wv = 0; wv < 8; ++wv) s += hist[wv][tid];
    atomicAdd(&gbins[tid], s);
  }
#endif
}

__global__ void ece_zero(float* gbins) {
  if (threadIdx.x < 48) gbins[threadIdx.x] = 0.0f;
}

__global__ void ece_final(const float* __restrict__ gbins,
                          float* __restrict__ out, float invN) {
  if (threadIdx.x == 0) {
    float e = 0.0f;
#pragma unroll
    for (int b = 0; b < NBINS; ++b) {
      float cnt = gbins[b];
      if (cnt > 0.0f) e += fabsf(gbins[16 + b] - gbins[32 + b]);
    }
    out[0] = e * invN;   // Σ |conf_sum - acc_sum| / N  ==  Σ gap*prop
  }
}

extern "C" void kernel_launch(void* const* d_in, const int* in_sizes, int n_in,
                              void* d_out, int out_size, void* d_ws, size_t ws_size,
                              hipStream_t stream) {
  const float* logits = (const float*)d_in[0];
  const int*   lab32  = (const int*)d_in[1];     // int64 labels, low dwords
  float*       gbins  = (float*)d_ws;
  int nElem  = in_sizes[1];                      // N (= in_sizes[0]/3)
  int nTiles = (nElem + TILE - 1) / TILE;
  int grid   = GRID < nTiles ? GRID : nTiles;

  ece_zero <<<1, 64, 0, stream>>>(gbins);
  ece_hist <<<grid, BLOCK, 0, stream>>>(logits, lab32, gbins, nElem, nTiles);
  ece_final<<<1, 32, 0, stream>>>(gbins, (float*)d_out, 1.0f / (float)nElem);
}